// L1Attn_41214506173057
// MI455X (gfx1250) — compile-verified
//
#include <hip/hip_runtime.h>
#include <stdint.h>

// L1 attention: out[b,i,j,h] = -(1/8) * sum_d |q[b,i,h,d] - k[b,j,h,d]|
// B=2, T=512, H=8, D=64.  VALU-bound kernel with CDNA5 async global->LDS staging.

namespace {
constexpr int Bc  = 2;
constexpr int Tt  = 512;
constexpr int Hh  = 8;
constexpr int Dd  = 64;
constexpr int TI  = 32;   // i-tile rows per block
constexpr int TJ  = 64;   // j-tile rows per block
constexpr int PAD = 68;   // floats per LDS row: lane stride 68 words -> bank 4*lane, conflict-free b128 reads
}

// CDNA5 async copy: 16 bytes global -> LDS per enabled lane, tracked by ASYNCcnt.
__device__ __forceinline__ void async_ld16(uint32_t lds_byte_off, const void* gptr) {
  asm volatile("global_load_async_to_lds_b128 %0, %1, off"
               :: "v"(lds_byte_off), "v"((uint64_t)(uintptr_t)gptr)
               : "memory");
}

__global__ __launch_bounds__(256)
void l1attn_tile(const float* __restrict__ q, const float* __restrict__ k,
                 float* __restrict__ out)
{
  __shared__ __align__(16) float sq[TI * PAD];
  __shared__ __align__(16) float sk[TJ * PAD];

  const int tid = threadIdx.x;
  const int jt  = blockIdx.x;        // 0..7   (j tiles of 64)
  const int it  = blockIdx.y;        // 0..15  (i tiles of 32)
  const int b   = blockIdx.z >> 3;   // 0..1
  const int h   = blockIdx.z & 7;    // 0..7

  // q/k layout (B,T,H,D): t-stride = H*D = 512 floats
  const float* qbase = q + (((size_t)b * Tt + (size_t)it * TI) * Hh + h) * Dd;
  const float* kbase = k + (((size_t)b * Tt + (size_t)jt * TJ) * Hh + h) * Dd;

  // LDS byte offsets (flat LDS addr low 32 bits == wave-relative LDS offset)
  const uint32_t sq0 = (uint32_t)(uintptr_t)(void*)sq;
  const uint32_t sk0 = (uint32_t)(uintptr_t)(void*)sk;

  // ---- stage q tile: 32 rows x 16 float4 = 512 chunks, 2 per thread ----
#pragma unroll
  for (int s = 0; s < 2; ++s) {
    const int chunk = tid + 256 * s;
    const int row = chunk >> 4;
    const int c4  = (chunk & 15) * 4;
    async_ld16(sq0 + (uint32_t)(row * PAD + c4) * 4u,
               qbase + (size_t)row * (Hh * Dd) + c4);
  }
  // ---- stage k tile: 64 rows x 16 float4 = 1024 chunks, 4 per thread ----
#pragma unroll
  for (int s = 0; s < 4; ++s) {
    const int chunk = tid + 256 * s;
    const int row = chunk >> 4;
    const int c4  = (chunk & 15) * 4;
    async_ld16(sk0 + (uint32_t)(row * PAD + c4) * 4u,
               kbase + (size_t)row * (Hh * Dd) + c4);
  }

  asm volatile("s_wait_asynccnt 0" ::: "memory");
  __syncthreads();

  // Micro-tile: 2 i-rows x 4 j-rows per thread.
  // jc in 0..15, j = jc + 16*m  -> k-tile lane stride = 68 words (bank 4*lane, conflict free)
  // i0 = 2*(tid>>4)             -> q reads are half-wave broadcasts
  const int jc = tid & 15;
  const int i0 = (tid >> 4) * 2;

  float acc[2][4];
#pragma unroll
  for (int n = 0; n < 2; ++n)
#pragma unroll
    for (int m = 0; m < 4; ++m) acc[n][m] = 0.0f;

#pragma unroll 4
  for (int d = 0; d < Dd; d += 4) {
    const float4 qa = *(const float4*)&sq[(i0 + 0) * PAD + d];
    const float4 qb = *(const float4*)&sq[(i0 + 1) * PAD + d];
    float4 kv[4];
#pragma unroll
    for (int m = 0; m < 4; ++m)
      kv[m] = *(const float4*)&sk[(jc + 16 * m) * PAD + d];

#pragma unroll
    for (int m = 0; m < 4; ++m) {
      // v_sub_f32 + v_add_f32 with |src| modifier: 2 VALU ops per (i,j,d)
      acc[0][m] += fabsf(qa.x - kv[m].x) + fabsf(qa.y - kv[m].y)
                 + fabsf(qa.z - kv[m].z) + fabsf(qa.w - kv[m].w);
      acc[1][m] += fabsf(qb.x - kv[m].x) + fabsf(qb.y - kv[m].y)
                 + fabsf(qb.z - kv[m].z) + fabsf(qb.w - kv[m].w);
    }
  }

  // out (B,T,T,H): i stride = T*H, j stride = H
  const float nscale = -0.125f;  // -1/sqrt(64)
  const size_t obase = (((size_t)b * Tt + (size_t)it * TI + i0) * Tt
                        + (size_t)jt * TJ + jc) * Hh + h;
#pragma unroll
  for (int n = 0; n < 2; ++n)
#pragma unroll
    for (int m = 0; m < 4; ++m)
      out[obase + (size_t)n * (Tt * Hh) + (size_t)(16 * m) * Hh] = nscale * acc[n][m];
}

extern "C" void kernel_launch(void* const* d_in, const int* in_sizes, int n_in,
                              void* d_out, int out_size, void* d_ws, size_t ws_size,
                              hipStream_t stream) {
  (void)in_sizes; (void)n_in; (void)out_size; (void)d_ws; (void)ws_size;
  const float* q = (const float*)d_in[0];
  const float* k = (const float*)d_in[1];
  float* out = (float*)d_out;

  dim3 grid(Tt / TJ, Tt / TI, Bc * Hh);  // (8, 16, 16) = 2048 blocks
  dim3 block(256);                       // 8 waves (wave32)
  l1attn_tile<<<grid, block, 0, stream>>>(q, k, out);
}